// Transformer_38706245271722
// MI455X (gfx1250) — compile-verified
//
#include <hip/hip_runtime.h>
#include <math.h>

#define NKEY 7
#define REP  12
#define MB   64
#define FF   256
#define BB   4
#define SS   1024
#define HH   8
#define LL   4
#define CC   512
#define NTOK (BB*SS)     // 4096
#define FEAT 96          // 12 * MB / HH

typedef float v2f __attribute__((ext_vector_type(2)));
typedef float v8f __attribute__((ext_vector_type(8)));

__constant__ int c_rowkey[REP]   = {0,1,2,2,3,3,4,4,5,5,6,6};
__constant__ int c_keybase[NKEY] = {0,1,2,4,6,8,10};
__constant__ int c_keyd[NKEY]    = {1,1,2,2,2,2,2};

// cos/sin at multiples of 30 degrees (exact D12 Fourier basis values)
__constant__ float c_cos12[12] = {1.f, 0.8660254037844387f, 0.5f, 0.f,
                                  -0.5f, -0.8660254037844387f, -1.f,
                                  -0.8660254037844387f, -0.5f, 0.f,
                                  0.5f, 0.8660254037844387f};
__constant__ float c_sin12[12] = {0.f, 0.5f, 0.8660254037844387f, 1.f,
                                  0.8660254037844387f, 0.5f, 0.f,
                                  -0.5f, -0.8660254037844387f, -1.f,
                                  -0.8660254037844387f, -0.5f};

// Row r, column n of the stacked 12x12 D12 Fourier basis Q.
__device__ __forceinline__ float q_entry(int r, int n) {
  const float inv_s12 = 0.28867513459481287f;  // 1/sqrt(12)
  const float inv_s6  = 0.40824829046386307f;  // 1/sqrt(6)
  if (r == 0) return inv_s12;
  if (r == 1) return (n & 1) ? -inv_s12 : inv_s12;
  int a = ((r >> 1) * n) % 12;                 // angle index, 30-degree units
  return inv_s6 * ((r & 1) ? c_sin12[a] : c_cos12[a]);
}

__device__ __forceinline__ float gelu_exact(float x) {
  return 0.5f * x * (1.0f + erff(x * 0.7071067811865476f));
}

__device__ __forceinline__ v8f wmma4(v2f a, v2f b, v8f c) {
  return __builtin_amdgcn_wmma_f32_16x16x4_f32(false, a, false, b, (short)0, c,
                                               false, false);
}

// ---------------------------------------------------------------------------
// Per-rep-row linear: Y[t][r][0:Nout] = X[t][r][0:Min] @ W[key(r)] (+ Res)
// One wave computes 32 tokens x 64 cols (8 tiles, 64 accumulator VGPRs).
// Compile-time Min/Nout -> pure marching-pointer addressing, no 64-bit muls.
// grid = (NTOK/32, Nout/64, 12), block = 32
// ---------------------------------------------------------------------------
template <int Min, int Nout>
__global__ __launch_bounds__(32, 1)
void k_rep_lin(const float* __restrict__ X, const float* __restrict__ W,
               float* __restrict__ Y, const float* __restrict__ Res) {
  const int t0    = blockIdx.x * 32;
  const int nbase = blockIdx.y * 64;
  const int r     = blockIdx.z;
  const int key   = c_rowkey[r];
  const int lane = threadIdx.x, half = lane >> 4, lr = lane & 15;

  const float* pa0 = X + ((t0 + lr) * REP + r) * Min + 2 * half;
  const float* pa1 = pa0 + 16 * REP * Min;
  const float* pb  = W + key * (Min * Nout) + (2 * half) * Nout + nbase + lr;

  v8f acc[2][4];
#pragma unroll
  for (int ti = 0; ti < 2; ++ti)
#pragma unroll
    for (int j = 0; j < 4; ++j)
#pragma unroll
      for (int e = 0; e < 8; ++e) acc[ti][j][e] = 0.0f;

  for (int k0 = 0; k0 < Min; k0 += 4) {
    v2f a0, a1, b[4];
    a0[0] = pa0[0]; a0[1] = pa0[1];
    a1[0] = pa1[0]; a1[1] = pa1[1];
#pragma unroll
    for (int j = 0; j < 4; ++j) { b[j][0] = pb[j * 16]; b[j][1] = pb[j * 16 + Nout]; }
#pragma unroll
    for (int j = 0; j < 4; ++j) {
      acc[0][j] = wmma4(a0, b[j], acc[0][j]);
      acc[1][j] = wmma4(a1, b[j], acc[1][j]);
    }
    pa0 += 4; pa1 += 4; pb += 4 * Nout;
  }

  const int obase = (t0 * REP + r) * Nout + nbase;
  float* yb = Y + obase;
  const float* rb = Res ? (Res + obase) : nullptr;
#pragma unroll
  for (int ti = 0; ti < 2; ++ti)
#pragma unroll
    for (int j = 0; j < 4; ++j)
#pragma unroll
      for (int e = 0; e < 8; ++e) {
        int off = (ti * 16 + e + 8 * half) * (REP * Nout) + j * 16 + lr;
        float v = acc[ti][j][e];
        if (rb) v += rb[off];
        yb[off] = v;
      }
}

// ---------------------------------------------------------------------------
// scores[z=(b,h)][s][s'] = (Q.K^T)/sqrt(96) with mask.
// One wave computes 32 queries x 64 keys (8 tiles).
// grid = (SS/32, SS/64, BB*HH), block = 32
// ---------------------------------------------------------------------------
__global__ __launch_bounds__(32, 1)
void k_attn_scores(const float* __restrict__ Q, const float* __restrict__ K,
                   float* __restrict__ Sc, const float* __restrict__ maskKey,
                   const float* __restrict__ maskQry, int causal) {
  const int s0 = blockIdx.x * 32;
  const int n0 = blockIdx.y * 64;
  const int z = blockIdx.z;
  const int b = z / HH, h = z % HH;
  const int lane = threadIdx.x, half = lane >> 4, lr = lane & 15;

  const float* q0 = Q + ((b * SS + s0 + lr) * REP) * MB + h * 8 + 2 * half;
  const float* q1 = q0 + 16 * REP * MB;
  const float* kb[4];
#pragma unroll
  for (int j = 0; j < 4; ++j)
    kb[j] = K + ((b * SS + n0 + j * 16 + lr) * REP) * MB + h * 8 + 2 * half;

  v8f acc[2][4];
#pragma unroll
  for (int ti = 0; ti < 2; ++ti)
#pragma unroll
    for (int j = 0; j < 4; ++j)
#pragma unroll
      for (int e = 0; e < 8; ++e) acc[ti][j][e] = 0.0f;

#pragma unroll 4
  for (int step = 0; step < FEAT / 4; ++step) {
    int off = (step >> 1) * MB + (step & 1) * 4;   // + folded 2*half in base
    v2f a0, a1, bv[4];
    a0[0] = q0[off]; a0[1] = q0[off + 1];
    a1[0] = q1[off]; a1[1] = q1[off + 1];
#pragma unroll
    for (int j = 0; j < 4; ++j) { bv[j][0] = kb[j][off]; bv[j][1] = kb[j][off + 1]; }
#pragma unroll
    for (int j = 0; j < 4; ++j) {
      acc[0][j] = wmma4(a0, bv[j], acc[0][j]);
      acc[1][j] = wmma4(a1, bv[j], acc[1][j]);
    }
  }

  const float scale = 0.10206207261596577f;  // 1/sqrt(96)
  float* sb = Sc + (size_t)z * SS * SS;
#pragma unroll
  for (int ti = 0; ti < 2; ++ti)
#pragma unroll
    for (int j = 0; j < 4; ++j)
#pragma unroll
      for (int e = 0; e < 8; ++e) {
        int qs = s0 + ti * 16 + e + 8 * half;
        int ks = n0 + j * 16 + lr;
        bool valid;
        if (causal) valid = (maskQry[b * SS + qs] > 0.5f) && (ks <= qs);
        else        valid = (maskKey[b * SS + ks] > 0.5f);
        float v = valid ? acc[ti][j][e] * scale : -1e9f;
        sb[qs * SS + ks] = v;
      }
}

// Row softmax over SS columns, one 256-thread block per row, in place.
__global__ void k_softmax(float* __restrict__ Sc) {
  __shared__ float red[256];
  const size_t row = blockIdx.x;
  float* P = Sc + row * (size_t)SS;
  const int tid = threadIdx.x;

  float mx = -3.4e38f;
  for (int j = tid; j < SS; j += 256) mx = fmaxf(mx, P[j]);
  red[tid] = mx; __syncthreads();
  for (int s = 128; s > 0; s >>= 1) { if (tid < s) red[tid] = fmaxf(red[tid], red[tid + s]); __syncthreads(); }
  mx = red[0]; __syncthreads();

  float sum = 0.0f;
  for (int j = tid; j < SS; j += 256) { float e = expf(P[j] - mx); P[j] = e; sum += e; }
  red[tid] = sum; __syncthreads();
  for (int s = 128; s > 0; s >>= 1) { if (tid < s) red[tid] += red[tid + s]; __syncthreads(); }
  float inv = 1.0f / red[0];
  for (int j = tid; j < SS; j += 256) P[j] *= inv;
}

// O[t=(b,qs)][r][h*8+u] = sum_{s'} P[b,h,qs,s'] * V[(b,s')][r][h*8+u]
// One wave computes 32 queries x 96 features (12 tiles).
// grid = (SS/32, 1, BB*HH), block = 32
__global__ __launch_bounds__(32, 1)
void k_attn_out(const float* __restrict__ P, const float* __restrict__ V,
                float* __restrict__ O) {
  const int s0 = blockIdx.x * 32;
  const int z = blockIdx.z;
  const int b = z / HH, h = z % HH;
  const int lane = threadIdx.x, half = lane >> 4, lr = lane & 15;

  const float* p0 = P + ((size_t)z * SS + (s0 + lr)) * SS + 2 * half;
  const float* p1 = p0 + 16 * SS;

  int noff[6];
#pragma unroll
  for (int j = 0; j < 6; ++j) {
    int n = j * 16 + lr;                       // mapped feature 0..95
    noff[j] = (n >> 3) * MB + (n & 7);
  }
  const float* vb = V + ((b * SS + 2 * half) * REP) * MB + h * 8;

  v8f acc[2][6];
#pragma unroll
  for (int ti = 0; ti < 2; ++ti)
#pragma unroll
    for (int j = 0; j < 6; ++j)
#pragma unroll
      for (int e = 0; e < 8; ++e) acc[ti][j][e] = 0.0f;

  for (int k0 = 0; k0 < SS; k0 += 4) {
    v2f a0, a1, bv[6];
    a0[0] = p0[0]; a0[1] = p0[1];
    a1[0] = p1[0]; a1[1] = p1[1];
#pragma unroll
    for (int j = 0; j < 6; ++j) { bv[j][0] = vb[noff[j]]; bv[j][1] = vb[noff[j] + REP * MB]; }
#pragma unroll
    for (int j = 0; j < 6; ++j) {
      acc[0][j] = wmma4(a0, bv[j], acc[0][j]);
      acc[1][j] = wmma4(a1, bv[j], acc[1][j]);
    }
    p0 += 4; p1 += 4; vb += 4 * REP * MB;
  }

  float* ob = O + ((b * SS + s0) * REP) * MB + h * 8;
#pragma unroll
  for (int ti = 0; ti < 2; ++ti)
#pragma unroll
    for (int j = 0; j < 6; ++j)
#pragma unroll
      for (int e = 0; e < 8; ++e) {
        int off = (ti * 16 + e + 8 * half) * (REP * MB) + noff[j];
        ob[off] = acc[ti][j][e];
      }
}

// --------------------------- elementwise kernels ---------------------------
__global__ void k_mask(const float* __restrict__ X, float* __restrict__ Msk, int inner) {
  int i = blockIdx.x * blockDim.x + threadIdx.x;
  if (i >= NTOK) return;
  const float* p = X + (size_t)i * inner;
  float s = 0.0f;
  for (int j = 0; j < inner; ++j) s += p[j];
  Msk[i] = (s > 0.0f) ? 1.0f : 0.0f;
}

__global__ void k_t12(const float* __restrict__ tgt, const float* __restrict__ Win,
                      const float* __restrict__ bin, float* __restrict__ t12) {
  int i = blockIdx.x * blockDim.x + threadIdx.x;
  if (i >= NTOK * REP) return;
  int n = i % REP, t = i / REP;
  const float* tp = tgt + (size_t)t * CC;
  float acc = bin[n];
  for (int c = 0; c < CC; ++c) acc += tp[c] * Win[(size_t)c * REP + n];
  t12[i] = acc;
}

// featurize + first (1 -> M) linear:  X[t][r][m] = t_r * W1[key(r)][m]
__global__ void k_feat_lin1(const float* __restrict__ F12, const float* __restrict__ bias,
                            const float* __restrict__ wgt, const float* __restrict__ W1,
                            float* __restrict__ X) {
  int i = blockIdx.x * blockDim.x + threadIdx.x;
  if (i >= NTOK * REP) return;
  int r = i % REP, t = i / REP;
  float bv = bias[0], wv = wgt[0];
  const float* fp = F12 + (size_t)t * REP;
  float tv = 0.0f;
  for (int n = 0; n < REP; ++n) tv += q_entry(r, n) * (fp[n] + bv);
  tv *= wv;
  const float* w1 = W1 + (size_t)c_rowkey[r] * MB;
  float* out = X + (size_t)i * MB;
  for (int m = 0; m < MB; ++m) out[m] = tv * w1[m];
}

// per-key basis gelu: x[i] <- QL[i] @ gelu(QL[i]^T @ x[i]), in place.
__global__ void k_act(float* __restrict__ X, int cols) {
  int i = blockIdx.x * blockDim.x + threadIdx.x;
  int total = NTOK * NKEY * cols;
  if (i >= total) return;
  int m = i % cols, k = (i / cols) % NKEY, t = i / (cols * NKEY);
  int d = c_keyd[k], rb = c_keybase[k];
  size_t base = ((size_t)t * REP + rb) * cols + m;
  float x0 = X[base];
  float x1 = (d == 2) ? X[base + cols] : 0.0f;
  float y0 = 0.0f, y1 = 0.0f;
  for (int n = 0; n < REP; ++n) {
    float q0 = q_entry(rb, n);
    float q1 = (d == 2) ? q_entry(rb + 1, n) : 0.0f;
    float g = gelu_exact(q0 * x0 + q1 * x1);
    y0 += q0 * g; y1 += q1 * g;
  }
  X[base] = y0;
  if (d == 2) X[base + cols] = y1;
}

// Only rep-row 0 receives positional encoding (other Q rows sum to zero).
__global__ void k_posenc(float* __restrict__ X) {
  int i = blockIdx.x * blockDim.x + threadIdx.x;
  if (i >= NTOK * MB) return;
  int m = i % MB, t = i / MB;
  int s = t % SS;
  float freq = expf(-(float)(m & ~1) * (9.210340371976184f / (float)MB));
  float v = (m & 1) ? cosf((float)s * freq) : sinf((float)s * freq);
  X[((size_t)t * REP) * MB + m] += 3.4641016151377544f * v;  // sqrt(12)
}

__global__ void k_fc(const float* __restrict__ X, const float* __restrict__ fw,
                     float* __restrict__ Y) {
  int i = blockIdx.x * blockDim.x + threadIdx.x;
  if (i >= NTOK * REP) return;
  int r = i % REP;
  const float* xp = X + (size_t)i * MB;
  const float* wp = fw + (size_t)c_rowkey[r] * MB;
  float acc = 0.0f;
  for (int m = 0; m < MB; ++m) acc += xp[m] * wp[m];
  Y[i] = acc;
}

__global__ void k_out12(const float* __restrict__ Y, float* __restrict__ O12) {
  int i = blockIdx.x * blockDim.x + threadIdx.x;
  if (i >= NTOK * REP) return;
  int n = i % REP, t = i / REP;
  const float* yp = Y + (size_t)t * REP;
  float acc = 0.0f;
  for (int r = 0; r < REP; ++r) acc += q_entry(r, n) * yp[r];
  O12[i] = acc * (1.0f / 7.0f);
}

__global__ void k_final(const float* __restrict__ O12, const float* __restrict__ Wout,
                        const float* __restrict__ bout, float* __restrict__ out) {
  int i = blockIdx.x * blockDim.x + threadIdx.x;
  if (i >= NTOK * CC) return;
  int c = i % CC, t = i / CC;
  const float* op = O12 + (size_t)t * REP;
  float acc = bout[c];
  for (int n = 0; n < REP; ++n) acc += op[n] * Wout[(size_t)n * CC + c];
  out[i] = acc;
}

// ---------------------------------------------------------------------------
extern "C" void kernel_launch(void* const* d_in, const int* in_sizes, int n_in,
                              void* d_out, int out_size, void* d_ws, size_t ws_size,
                              hipStream_t stream) {
  (void)in_sizes; (void)n_in; (void)out_size; (void)ws_size;
  const float* src     = (const float*)d_in[0];
  const float* tgt     = (const float*)d_in[1];
  const float* enc_fb  = (const float*)d_in[2];
  const float* enc_fw  = (const float*)d_in[3];
  const float* dec_fb  = (const float*)d_in[4];
  const float* dec_fw  = (const float*)d_in[5];
  const float* enc_w1  = (const float*)d_in[6];
  const float* enc_w2  = (const float*)d_in[7];
  const float* enc_w3  = (const float*)d_in[8];
  const float* dec_w1  = (const float*)d_in[9];
  const float* dec_w2  = (const float*)d_in[10];
  const float* dec_w3  = (const float*)d_in[11];
  const float* enc_wq  = (const float*)d_in[12];
  const float* enc_wk  = (const float*)d_in[13];
  const float* enc_wv  = (const float*)d_in[14];
  const float* enc_wo  = (const float*)d_in[15];
  const float* enc_ff1 = (const float*)d_in[16];
  const float* enc_ff2 = (const float*)d_in[17];
  const float* dec_swq = (const float*)d_in[18];
  const float* dec_swk = (const float*)d_in[19];
  const float* dec_swv = (const float*)d_in[20];
  const float* dec_swo = (const float*)d_in[21];
  const float* dec_cwq = (const float*)d_in[22];
  const float* dec_cwk = (const float*)d_in[23];
  const float* dec_cwv = (const float*)d_in[24];
  const float* dec_cwo = (const float*)d_in[25];
  const float* dec_ff1 = (const float*)d_in[26];
  const float* dec_ff2 = (const float*)d_in[27];
  const float* fc_w    = (const float*)d_in[28];
  const float* vq_in_w = (const float*)d_in[29];
  const float* vq_in_b = (const float*)d_in[30];
  const float* vq_out_w= (const float*)d_in[31];
  const float* vq_out_b= (const float*)d_in[32];

  const size_t XSZ = (size_t)NTOK * REP * MB;
  float* ws    = (float*)d_ws;
  float* encX  = ws;
  float* decX  = encX + XSZ;
  float* bufQ  = decX + XSZ;
  float* bufK  = bufQ + XSZ;
  float* bufV  = bufK + XSZ;
  float* bufT  = bufV + XSZ;
  float* bufH  = bufT + XSZ;                               // NTOK*REP*FF
  float* scores= bufH + (size_t)NTOK * REP * FF;           // BB*HH*SS*SS
  float* t12   = scores + (size_t)BB * HH * SS * SS;
  float* yrep  = t12 + (size_t)NTOK * REP;
  float* o12   = yrep + (size_t)NTOK * REP;
  float* maskS = o12 + (size_t)NTOK * REP;
  float* maskT = maskS + NTOK;

  const dim3 B256(256);
  const dim3 WV(32);
  const dim3 gRep((NTOK * REP + 255) / 256);
  const dim3 gLin(NTOK / 32, 1, REP);

  // masks + decoder 12-dim input
  k_mask<<<dim3((NTOK + 255) / 256), B256, 0, stream>>>(src, maskS, REP);
  k_mask<<<dim3((NTOK + 255) / 256), B256, 0, stream>>>(tgt, maskT, CC);
  k_t12<<<gRep, B256, 0, stream>>>(tgt, vq_in_w, vq_in_b, t12);

  auto embed = [&](const float* F12, const float* fb, const float* fw,
                   const float* W1, const float* W2, const float* W3, float* X) {
    k_feat_lin1<<<gRep, B256, 0, stream>>>(F12, fb, fw, W1, X);
    k_act<<<dim3((NTOK * NKEY * MB + 255) / 256), B256, 0, stream>>>(X, MB);
    k_rep_lin<MB, MB><<<gLin, WV, 0, stream>>>(X, W2, bufT, nullptr);
    k_act<<<dim3((NTOK * NKEY * MB + 255) / 256), B256, 0, stream>>>(bufT, MB);
    k_rep_lin<MB, MB><<<gLin, WV, 0, stream>>>(bufT, W3, X, nullptr);
    k_posenc<<<dim3((NTOK * MB + 255) / 256), B256, 0, stream>>>(X);
  };
  embed(src, enc_fb, enc_fw, enc_w1, enc_w2, enc_w3, encX);
  embed(t12, dec_fb, dec_fw, dec_w1, dec_w2, dec_w3, decX);

  auto mha = [&](float* Xq, float* Xkv, const float* Wq, const float* Wk,
                 const float* Wv, const float* Wo, const float* mKey,
                 const float* mQry, int causal) {
    k_rep_lin<MB, MB><<<gLin, WV, 0, stream>>>(Xq,  Wq, bufQ, nullptr);
    k_rep_lin<MB, MB><<<gLin, WV, 0, stream>>>(Xkv, Wk, bufK, nullptr);
    k_rep_lin<MB, MB><<<gLin, WV, 0, stream>>>(Xkv, Wv, bufV, nullptr);
    k_attn_scores<<<dim3(SS / 32, SS / 64, BB * HH), WV, 0, stream>>>(
        bufQ, bufK, scores, mKey, mQry, causal);
    k_softmax<<<dim3(BB * HH * SS), B256, 0, stream>>>(scores);
    k_attn_out<<<dim3(SS / 32, 1, BB * HH), WV, 0, stream>>>(scores, bufV, bufT);
    k_rep_lin<MB, MB><<<gLin, WV, 0, stream>>>(bufT, Wo, Xq, Xq);  // +residual
  };
  auto ffn = [&](float* X, const float* W1, const float* W2) {
    k_rep_lin<MB, FF><<<dim3(NTOK / 32, FF / 64, REP), WV, 0, stream>>>(X, W1, bufH, nullptr);
    k_act<<<dim3((NTOK * NKEY * FF + 255) / 256), B256, 0, stream>>>(bufH, FF);
    k_rep_lin<FF, MB><<<gLin, WV, 0, stream>>>(bufH, W2, X, X);
  };

  const size_t WMM = (size_t)NKEY * MB * MB;
  const size_t WF1 = (size_t)NKEY * MB * FF;
  const size_t WF2 = (size_t)NKEY * FF * MB;

  for (int l = 0; l < LL; ++l) {
    mha(encX, encX, enc_wq + l * WMM, enc_wk + l * WMM, enc_wv + l * WMM,
        enc_wo + l * WMM, maskS, maskS, 0);
    ffn(encX, enc_ff1 + l * WF1, enc_ff2 + l * WF2);
  }
  for (int l = 0; l < LL; ++l) {
    mha(decX, decX, dec_swq + l * WMM, dec_swk + l * WMM, dec_swv + l * WMM,
        dec_swo + l * WMM, maskT, maskT, 1);
    mha(decX, encX, dec_cwq + l * WMM, dec_cwk + l * WMM, dec_cwv + l * WMM,
        dec_cwo + l * WMM, maskS, maskS, 0);
    ffn(decX, dec_ff1 + l * WF1, dec_ff2 + l * WF2);
  }

  k_fc<<<gRep, B256, 0, stream>>>(decX, fc_w, yrep);
  k_out12<<<gRep, B256, 0, stream>>>(yrep, o12);
  k_final<<<dim3((NTOK * CC + 255) / 256), B256, 0, stream>>>(o12, vq_out_w, vq_out_b,
                                                              (float*)d_out);
}